// ResGatedGCN_28836410425876
// MI455X (gfx1250) — compile-verified
//
#include <hip/hip_runtime.h>
#include <hip/hip_bf16.h>
#include <math.h>

typedef __attribute__((ext_vector_type(2))) float v2f;
typedef __attribute__((ext_vector_type(8))) float v8f;

#define HIDC 128
#define LDSTRIDE 34   // 34 floats = 136 B: 8-byte aligned rows, spreads banks

// ---------------------------------------------------------------------------
// Fused 4-matrix GEMM: O = X @ W + b for one of 4 weight sets (blockIdx.y).
// Block: 256 threads (8 waves). Each block -> 128 rows x 128 cols.
// Each wave owns a 16-row M tile and all 8 N tiles (8 x v8f accumulators).
// fp32 WMMA 16x16x4, K staged in LDS in chunks of 32.
// ---------------------------------------------------------------------------
__global__ __launch_bounds__(256) void resgcn_gemm4(
    const float* __restrict__ X,
    const float* __restrict__ W0, const float* __restrict__ W1,
    const float* __restrict__ W2, const float* __restrict__ W3,
    const float* __restrict__ B0, const float* __restrict__ B1,
    const float* __restrict__ B2, const float* __restrict__ B3,
    float* __restrict__ O0, float* __restrict__ O1,
    float* __restrict__ O2, float* __restrict__ O3,
    int n_rows)
{
    __shared__ float Xs[128 * LDSTRIDE];
    __shared__ float Wt[128 * LDSTRIDE];

    const int tid  = threadIdx.x;
    const int wave = tid >> 5;
    const int lane = tid & 31;
    const int half = lane >> 4;   // 0: lanes 0-15, 1: lanes 16-31
    const int l16  = lane & 15;
    const int row0 = blockIdx.x * 128;

    const float* W; const float* bias; float* O;
    switch (blockIdx.y) {
        case 0:  W = W0; bias = B0; O = O0; break;
        case 1:  W = W1; bias = B1; O = O1; break;
        case 2:  W = W2; bias = B2; O = O2; break;
        default: W = W3; bias = B3; O = O3; break;
    }

    v8f acc[8];
    #pragma unroll
    for (int i = 0; i < 8; i++)
        #pragma unroll
        for (int j = 0; j < 8; j++) acc[i][j] = 0.0f;

    for (int kk = 0; kk < HIDC; kk += 32) {
        __syncthreads();
        // Stage X chunk: rows row0..row0+127, cols kk..kk+31 (clamped rows)
        #pragma unroll
        for (int p = 0; p < 4; p++) {
            int t  = p * 256 + tid;       // 0..1023
            int r  = t >> 3;              // 0..127
            int cs = (t & 7) << 2;        // 0,4,...,28
            int grow = row0 + r;
            if (grow >= n_rows) grow = n_rows - 1;
            float4 val = *(const float4*)(X + (size_t)grow * HIDC + kk + cs);
            float* dp = &Xs[r * LDSTRIDE + cs];
            dp[0] = val.x; dp[1] = val.y; dp[2] = val.z; dp[3] = val.w;
        }
        // Stage W chunk transposed: Wt[n][k_local], k_local = 0..31
        #pragma unroll
        for (int p = 0; p < 4; p++) {
            int t  = p * 256 + tid;
            int kr = t >> 5;              // 0..31
            int cs = (t & 31) << 2;       // 0..124
            float4 val = *(const float4*)(W + (size_t)(kk + kr) * HIDC + cs);
            Wt[(cs + 0) * LDSTRIDE + kr] = val.x;
            Wt[(cs + 1) * LDSTRIDE + kr] = val.y;
            Wt[(cs + 2) * LDSTRIDE + kr] = val.z;
            Wt[(cs + 3) * LDSTRIDE + kr] = val.w;
        }
        __syncthreads();

        const int m = wave * 16 + l16;
        #pragma unroll
        for (int ks = 0; ks < 32; ks += 4) {
            // A frag (16x4 f32): lanes 0-15 hold K=ks,ks+1; lanes 16-31 K=ks+2,ks+3
            v2f a = *(const v2f*)&Xs[m * LDSTRIDE + ks + 2 * half];
            #pragma unroll
            for (int nt = 0; nt < 8; nt++) {
                int n = nt * 16 + l16;
                v2f b = *(const v2f*)&Wt[n * LDSTRIDE + ks + 2 * half];
                acc[nt] = __builtin_amdgcn_wmma_f32_16x16x4_f32(
                    false, a, false, b, (short)0, acc[nt], false, false);
            }
        }
    }

    // Epilogue: D[r] holds rows (mbase + r + 8*half), col n = nt*16 + l16
    const int mbase = row0 + wave * 16;
    #pragma unroll
    for (int nt = 0; nt < 8; nt++) {
        int n = nt * 16 + l16;
        float bv = bias[n];
        #pragma unroll
        for (int r = 0; r < 8; r++) {
            int grow = mbase + r + 8 * half;
            if (grow < n_rows) O[(size_t)grow * HIDC + n] = acc[nt][r] + bv;
        }
    }
}

// ---------------------------------------------------------------------------
// Edge kernel: one wave per edge. gate = sigmoid(K[dst]+Q[src]);
// AGG[dst] += gate * V[src]  (AGG pre-initialized with X@Ws + bs by gemm4)
// ---------------------------------------------------------------------------
__global__ __launch_bounds__(256) void resgcn_edge(
    const int* __restrict__ esrc, const int* __restrict__ edst,
    const float* __restrict__ K, const float* __restrict__ Q,
    const float* __restrict__ V, float* __restrict__ AGG, int n_edges)
{
    int wave = threadIdx.x >> 5;
    int lane = threadIdx.x & 31;
    int e = blockIdx.x * 8 + wave;
    if (e >= n_edges) return;
    int s = esrc[e];
    int d = edst[e];
    int c = lane << 2;   // 4 channels per lane

    float4 kv = *(const float4*)(K + (size_t)d * HIDC + c);
    float4 qv = *(const float4*)(Q + (size_t)s * HIDC + c);
    float4 vv = *(const float4*)(V + (size_t)s * HIDC + c);

    float gx = 1.0f / (1.0f + expf(-(kv.x + qv.x)));
    float gy = 1.0f / (1.0f + expf(-(kv.y + qv.y)));
    float gz = 1.0f / (1.0f + expf(-(kv.z + qv.z)));
    float gw = 1.0f / (1.0f + expf(-(kv.w + qv.w)));

    float* ap = AGG + (size_t)d * HIDC + c;
    atomicAdd(ap + 0, gx * vv.x);
    atomicAdd(ap + 1, gy * vv.y);
    atomicAdd(ap + 2, gz * vv.z);
    atomicAdd(ap + 3, gw * vv.w);
}

// ---------------------------------------------------------------------------
// H = relu(AGG + b[channel])
// ---------------------------------------------------------------------------
__global__ __launch_bounds__(256) void resgcn_relu_bias(
    const float* __restrict__ AGG, const float* __restrict__ b,
    float* __restrict__ H, int total4)
{
    int i4 = blockIdx.x * 256 + threadIdx.x;
    if (i4 >= total4) return;
    int base = i4 << 2;
    int c = base & (HIDC - 1);
    float4 a = *(const float4*)(AGG + base);
    float4 r;
    r.x = fmaxf(a.x + b[c + 0], 0.0f);
    r.y = fmaxf(a.y + b[c + 1], 0.0f);
    r.z = fmaxf(a.z + b[c + 2], 0.0f);
    r.w = fmaxf(a.w + b[c + 3], 0.0f);
    *(float4*)(H + base) = r;
}

// ---------------------------------------------------------------------------
// Global mean pool: one block per graph; batch is sorted -> binary search.
// ---------------------------------------------------------------------------
__global__ __launch_bounds__(256) void resgcn_pool(
    const float* __restrict__ H, const int* __restrict__ batch,
    float* __restrict__ P, int n_nodes)
{
    int g = blockIdx.x;
    int lo = 0, hi = n_nodes;
    while (lo < hi) { int mid = (lo + hi) >> 1; if (batch[mid] < g) lo = mid + 1; else hi = mid; }
    int start = lo;
    hi = n_nodes;
    while (lo < hi) { int mid = (lo + hi) >> 1; if (batch[mid] < g + 1) lo = mid + 1; else hi = mid; }
    int end = lo;

    int c    = threadIdx.x & (HIDC - 1);
    int half = threadIdx.x >> 7;
    float sum = 0.0f;
    for (int i = start + half; i < end; i += 2)
        sum += H[(size_t)i * HIDC + c];

    __shared__ float red[256];
    red[threadIdx.x] = sum;
    __syncthreads();
    if (half == 0) {
        float tot = red[c] + red[HIDC + c];
        float cnt = (float)(end - start);
        P[g * HIDC + c] = tot / fmaxf(cnt, 1.0f);
    }
}

// ---------------------------------------------------------------------------
// Final FC: [64,128] @ [128,10] + b
// ---------------------------------------------------------------------------
__global__ __launch_bounds__(256) void resgcn_fc(
    const float* __restrict__ P, const float* __restrict__ Wf,
    const float* __restrict__ bf, float* __restrict__ out, int total)
{
    int idx = blockIdx.x * 256 + threadIdx.x;
    if (idx >= total) return;
    int g = idx / 10, j = idx % 10;
    float s = bf[j];
    #pragma unroll 4
    for (int c = 0; c < HIDC; c++) s += P[g * HIDC + c] * Wf[c * 10 + j];
    out[idx] = s;
}

extern "C" void kernel_launch(void* const* d_in, const int* in_sizes, int n_in,
                              void* d_out, int out_size, void* d_ws, size_t ws_size,
                              hipStream_t stream) {
    const float* x    = (const float*)d_in[0];
    const int*   edge = (const int*)d_in[1];
    const int*   batch= (const int*)d_in[2];
    const float* Wk1 = (const float*)d_in[3];  const float* bk1 = (const float*)d_in[4];
    const float* Wq1 = (const float*)d_in[5];  const float* bq1 = (const float*)d_in[6];
    const float* Wv1 = (const float*)d_in[7];  const float* bv1 = (const float*)d_in[8];
    const float* Ws1 = (const float*)d_in[9];  const float* bs1 = (const float*)d_in[10];
    const float* b1  = (const float*)d_in[11];
    const float* Wk2 = (const float*)d_in[12]; const float* bk2 = (const float*)d_in[13];
    const float* Wq2 = (const float*)d_in[14]; const float* bq2 = (const float*)d_in[15];
    const float* Wv2 = (const float*)d_in[16]; const float* bv2 = (const float*)d_in[17];
    const float* Ws2 = (const float*)d_in[18]; const float* bs2 = (const float*)d_in[19];
    const float* b2  = (const float*)d_in[20];
    const float* fcW = (const float*)d_in[21]; const float* fcb = (const float*)d_in[22];
    float* out = (float*)d_out;

    const int N = in_sizes[0] / HIDC;       // 50000
    const int E = in_sizes[1] / 2;          // 800000
    const size_t NC = (size_t)N * HIDC;

    float* ws  = (float*)d_ws;
    float* Kb  = ws;
    float* Qb  = ws + NC;
    float* Vb  = ws + 2 * NC;
    float* AGG = ws + 3 * NC;               // gets X@Ws + bs, then edge atomics
    float* H   = ws + 4 * NC;
    float* P   = ws + 5 * NC;               // 64*128 pooled

    const int* esrc = edge;
    const int* edst = edge + E;

    dim3 ggrid((N + 127) / 128, 4);
    int egrid = (E + 7) / 8;
    int rgrid = (int)((NC / 4 + 255) / 256);

    // ---- Layer 1 ----
    resgcn_gemm4<<<ggrid, 256, 0, stream>>>(x, Wk1, Wq1, Wv1, Ws1,
                                            bk1, bq1, bv1, bs1,
                                            Kb, Qb, Vb, AGG, N);
    resgcn_edge<<<egrid, 256, 0, stream>>>(esrc, edst, Kb, Qb, Vb, AGG, E);
    resgcn_relu_bias<<<rgrid, 256, 0, stream>>>(AGG, b1, H, (int)(NC / 4));

    // ---- Layer 2 ----
    resgcn_gemm4<<<ggrid, 256, 0, stream>>>(H, Wk2, Wq2, Wv2, Ws2,
                                            bk2, bq2, bv2, bs2,
                                            Kb, Qb, Vb, AGG, N);
    resgcn_edge<<<egrid, 256, 0, stream>>>(esrc, edst, Kb, Qb, Vb, AGG, E);
    resgcn_relu_bias<<<rgrid, 256, 0, stream>>>(AGG, b2, H, (int)(NC / 4));

    // ---- Pool + FC ----
    resgcn_pool<<<64, 256, 0, stream>>>(H, batch, P, N);
    resgcn_fc<<<3, 256, 0, stream>>>(P, fcW, fcb, out, 64 * 10);
}